// DecoderTF_39685497815523
// MI455X (gfx1250) — compile-verified
//
#include <hip/hip_runtime.h>

typedef __attribute__((ext_vector_type(2))) float v2f;
typedef __attribute__((ext_vector_type(8))) float v8f;

#define NN 512       // feature dim (GEMM reduction dim)
#define LL 16        // frame length
#define HOPW 8       // hop
#define MM 8
#define CC 2
#define KK 6000      // frames (GEMM M dim)
#define TT 48008     // (KK-1)*HOPW + LL
#define KTILE 16
#define NKT (KK / KTILE)                  // 375 k-tiles
#define TILE_T (HOPW * (KTILE - 1) + LL)  // 136 samples covered by one k-tile
#define WAVES 8                           // waves per block, each takes NN/WAVES n-rows
#define NCHUNK (NN / WAVES)               // 64 n-rows per wave

__global__ void zero_out_kernel(float* __restrict__ out, int n) {
    int i = blockIdx.x * blockDim.x + threadIdx.x;
    if (i < n) out[i] = 0.0f;
}

// 256-thread block (8 waves) per (m, k-tile); computes both c channels.
// Wave w reduces n in [64w, 64w+64); partial 16x16 D tiles from all waves
// are combined in the LDS overlap-add buffer via ds_add_f32.
__global__ __launch_bounds__(256) void decoder_oaa_wmma(
    const float* __restrict__ inputs,   // [M, N, K]
    const float* __restrict__ mask,     // [M, C, N, K]
    const float* __restrict__ W,        // [N, L]
    float* __restrict__ out) {          // [M, C, T]
    const int ktile = blockIdx.x;            // 0..374
    const int m     = blockIdx.y;            // 0..7
    const int tid   = threadIdx.x;           // 0..255
    const int lane  = tid & 31;              // lane in wave
    const int wave  = tid >> 5;              // 0..7
    const int half  = lane >> 4;             // lane-half selects GEMM-K pair
    const int kr    = lane & 15;             // row (A: k) / col (B: l) in half
    const int kb    = ktile * KTILE;

    // LDS staging for overlap-add of this tile (pad 136 -> 144 per channel).
    __shared__ float sacc[CC][TILE_T + 8];
    for (int i = tid; i < CC * (TILE_T + 8); i += 256)
        (&sacc[0][0])[i] = 0.0f;
    __syncthreads();

    // Per-lane base pointers at column (kb + kr); each wave offsets by its
    // n-chunk so the block streams the full N=512 reduction cooperatively.
    const int nbase = wave * NCHUNK;
    const size_t noff = (size_t)nbase * KK;
    const float* inp0 = inputs + (size_t)m * NN * KK          + noff + (size_t)(kb + kr);
    const float* msk0 = mask + ((size_t)m * CC + 0) * NN * KK + noff + (size_t)(kb + kr);
    const float* msk1 = mask + ((size_t)m * CC + 1) * NN * KK + noff + (size_t)(kb + kr);
    const float* Wb   = W + nbase * LL;

    v8f acc0 = {};  // c = 0 partial accumulator, D = 16x16 f32
    v8f acc1 = {};  // c = 1 partial accumulator

    for (int n0 = 0; n0 < NCHUNK; n0 += 4) {
        // A/B f32 layout: VGPR j, lane-half h holds GEMM-K index (2h + j)
        const int na = n0 + 2 * half;           // row for .x
        const size_t ra = (size_t)na * KK;      // row na
        const size_t rb = ra + KK;              // row na + 1 (for .y)

        float i0 = inp0[ra];
        float i1 = inp0[rb];
        float p0 = msk0[ra];
        float p1 = msk0[rb];
        float q0 = msk1[ra];
        float q1 = msk1[rb];

        v2f a0 = { i0 * p0, i1 * p1 };          // masked A fragment, c=0
        v2f a1 = { i0 * q0, i1 * q1 };          // masked A fragment, c=1
        v2f b  = { Wb[na * LL + kr], Wb[(na + 1) * LL + kr] };  // B = W[n, l]

        acc0 = __builtin_amdgcn_wmma_f32_16x16x4_f32(
            false, a0, false, b, (short)0, acc0, false, false);
        acc1 = __builtin_amdgcn_wmma_f32_16x16x4_f32(
            false, a1, false, b, (short)0, acc1, false, false);
    }

    // D layout: VGPR v, lane-half h -> frame row (v + 8h); col l = kr.
    // Overlap-add within the tile via LDS float atomics; this also performs
    // the cross-wave reduction of the 8 partial D tiles.
#pragma unroll
    for (int v = 0; v < 8; ++v) {
        const int krow = v + 8 * half;
        const int t = HOPW * krow + kr;   // 0..135
        atomicAdd(&sacc[0][t], acc0[v]);
        atomicAdd(&sacc[1][t], acc1[v]);
    }
    __syncthreads();

    // Flush tile to global; head/tail 8 samples overlap neighbor tiles ->
    // global_atomic_add_f32 resolves cross-workgroup accumulation.
    const size_t tbase = (size_t)ktile * (HOPW * KTILE);   // 128 per tile
    float* out0 = out + ((size_t)m * CC + 0) * TT + tbase;
    float* out1 = out + ((size_t)m * CC + 1) * TT + tbase;
    if (tid < TILE_T) {
        atomicAdd(&out0[tid], sacc[0][tid]);
        atomicAdd(&out1[tid], sacc[1][tid]);
    }
}

extern "C" void kernel_launch(void* const* d_in, const int* in_sizes, int n_in,
                              void* d_out, int out_size, void* d_ws, size_t ws_size,
                              hipStream_t stream) {
    const float* inputs   = (const float*)d_in[0];  // [8, 512, 6000]
    const float* est_mask = (const float*)d_in[1];  // [8, 2, 512, 6000]
    const float* W        = (const float*)d_in[2];  // [512, 16]
    float* out            = (float*)d_out;          // [8, 2, 48008]

    // d_out is poisoned by the harness; zero it before atomic accumulation.
    zero_out_kernel<<<(out_size + 255) / 256, 256, 0, stream>>>(out, out_size);

    dim3 grid(NKT, MM);   // (375, 8) blocks x 256 threads = 24000 waves
    decoder_oaa_wmma<<<grid, 256, 0, stream>>>(inputs, est_mask, W, out);
}